// AdaptiveEdgeWeightGNN_19447611916814
// MI455X (gfx1250) — compile-verified
//
#include <hip/hip_runtime.h>

#define N_NODES 50000
#define N_EDGES 800000
#define D 64

typedef float v2f __attribute__((ext_vector_type(2)));
typedef float v8f __attribute__((ext_vector_type(8)));

// ---------------------------------------------------------------------------
// 1) deg[n] = 1.0 (self loop)
__global__ void k_init_deg(float* __restrict__ deg) {
  int n = blockIdx.x * blockDim.x + threadIdx.x;
  if (n < N_NODES) deg[n] = 1.0f;
}

// 2) deg[col[e]] += ew[e]
__global__ void k_deg_scatter(const int* __restrict__ col, const float* __restrict__ ew,
                              float* __restrict__ deg) {
  int e = blockIdx.x * blockDim.x + threadIdx.x;
  if (e < N_EDGES) atomicAdd(&deg[col[e]], ew[e]);
}

// 3) dis[n] = rsqrt(deg[n])   (in place; deg >= 1 always, keep where for parity)
__global__ void k_rsqrt_inplace(float* __restrict__ deg) {
  int n = blockIdx.x * blockDim.x + threadIdx.x;
  if (n < N_NODES) {
    float d = deg[n];
    deg[n] = d > 0.0f ? rsqrtf(d) : 0.0f;
  }
}

// 4) xw = x @ W via V_WMMA_F32_16X16X4_F32. One wave per 16x16 output tile.
//    A 16x4 f32 layout : lane(0-15)=M, VGPR0/1 = K0/K1; lanes 16-31 hold K2/K3.
//    B 4x16 f32 layout : lane(0-15)=N, VGPR0/1 = K0/K1; lanes 16-31 hold K2/K3.
//    C/D 16x16 f32     : VGPR r -> M = r (+8 for lanes 16-31), N = lane&15.
__global__ void k_gemm_wmma(const float* __restrict__ x, const float* __restrict__ W,
                            float* __restrict__ xw) {
  const int lane = threadIdx.x & 31;
  const int wave = (blockIdx.x * blockDim.x + threadIdx.x) >> 5;
  const int tile_m = wave >> 2;      // D/16 = 4 tiles along output columns
  const int tile_n = wave & 3;
  if (tile_m * 16 >= N_NODES) return;   // wave-uniform branch (EXEC stays all-1s)

  const int half = lane >> 4;        // selects K-pair {0,1} vs {2,3}
  const int l    = lane & 15;
  const float* xrow = x + (size_t)(tile_m * 16 + l) * D;
  const int n_col = tile_n * 16 + l;

  v8f c = {};
#pragma unroll
  for (int k = 0; k < D; k += 4) {
    const int ka = k + half * 2;
    v2f a, b;
    a.x = xrow[ka];
    a.y = xrow[ka + 1];
    b.x = W[(size_t)ka * D + n_col];
    b.y = W[(size_t)(ka + 1) * D + n_col];
    // 8 args: (neg_a, A, neg_b, B, c_mod, C, reuse_a, reuse_b)
    c = __builtin_amdgcn_wmma_f32_16x16x4_f32(false, a, false, b, (short)0, c,
                                              false, false);
  }
#pragma unroll
  for (int r = 0; r < 8; ++r) {
    const int m = tile_m * 16 + half * 8 + r;
    xw[(size_t)m * D + n_col] = c[r];
  }
}

// 5) out[n,d] = bias[d] + dis[n]^2 * xw[n,d]   (self-loop contribution)
__global__ void k_init_out(const float* __restrict__ xw, const float* __restrict__ dis,
                           const float* __restrict__ bias, float* __restrict__ out) {
  int i = blockIdx.x * blockDim.x + threadIdx.x;
  if (i < N_NODES * D) {
    int n = i >> 6;
    int d = i & 63;
    float s = dis[n];
    out[i] = bias[d] + s * s * xw[i];
  }
}

// 6) out[col] += dis[row]*ew*dis[col] * xw[row]
//    16 threads per edge, float4 per thread: global_load_b128 gather (L2-resident
//    xw) + 4x global_atomic_add_f32 scatter (L2-resident out).
__global__ void k_edge_scatter(const int* __restrict__ row, const int* __restrict__ col,
                               const float* __restrict__ ew, const float* __restrict__ dis,
                               const float* __restrict__ xw, float* __restrict__ out) {
  long long i = (long long)blockIdx.x * blockDim.x + threadIdx.x;
  if (i >= (long long)N_EDGES * 16) return;
  int e = (int)(i >> 4);
  int t = (int)(i & 15);
  int r = row[e];
  int c = col[e];
  float norm = dis[r] * ew[e] * dis[c];
  const float4 v = *(const float4*)(xw + (size_t)r * D + t * 4);
  float* o = out + (size_t)c * D + t * 4;
  atomicAdd(o + 0, norm * v.x);
  atomicAdd(o + 1, norm * v.y);
  atomicAdd(o + 2, norm * v.z);
  atomicAdd(o + 3, norm * v.w);
}

// ---------------------------------------------------------------------------
extern "C" void kernel_launch(void* const* d_in, const int* in_sizes, int n_in,
                              void* d_out, int out_size, void* d_ws, size_t ws_size,
                              hipStream_t stream) {
  const float* x    = (const float*)d_in[0];   // [N, 64]
  const float* W    = (const float*)d_in[1];   // [64, 64]
  const float* bias = (const float*)d_in[2];   // [64]
  const float* ew   = (const float*)d_in[3];   // [E]
  const int*   ei   = (const int*)d_in[4];     // [2, E]
  const int* row = ei;
  const int* col = ei + N_EDGES;
  float* out = (float*)d_out;

  // workspace: deg/dis (N f32), then xw (N*64 f32, offset 200000 B = 16B aligned)
  float* deg = (float*)d_ws;
  float* xw  = deg + N_NODES;

  k_init_deg<<<(N_NODES + 255) / 256, 256, 0, stream>>>(deg);
  k_deg_scatter<<<(N_EDGES + 255) / 256, 256, 0, stream>>>(col, ew, deg);
  k_rsqrt_inplace<<<(N_NODES + 255) / 256, 256, 0, stream>>>(deg);

  const int tiles = (N_NODES / 16) * (D / 16);          // 12500 waves
  k_gemm_wmma<<<(tiles + 7) / 8, 256, 0, stream>>>(x, W, xw);  // 8 waves/block

  k_init_out<<<(N_NODES * D + 255) / 256, 256, 0, stream>>>(xw, deg, bias, out);

  long long nscatter = (long long)N_EDGES * 16;
  k_edge_scatter<<<(int)((nscatter + 255) / 256), 256, 0, stream>>>(row, col, ew, deg,
                                                                    xw, out);
}